// ML_Class_Capsule_35656818491738
// MI455X (gfx1250) — compile-verified
//
#include <hip/hip_runtime.h>
#include <math.h>

// Capsule dynamic routing (routings=2) for MI455X / gfx1250, wave32 + fp32 WMMA.
// pred is never materialized: 3 WMMA passes recompute it out of L2-resident W/X.
// Split-K over input capsules (i) gives 16x more waves + 16x shorter WMMA chains.

typedef float v2f __attribute__((ext_vector_type(2)));
typedef float v4f __attribute__((ext_vector_type(4)));
typedef float v8f __attribute__((ext_vector_type(8)));

#define NB 64
#define NI 1152
#define NK 8
#define ND 16
#define SPLITK 16
#define ICHUNK (NI / SPLITK) /* 72 */
#define EPSQ 1e-7f

// One 16x16 fp32 tile, contraction K=8 = two chained V_WMMA_F32_16X16X4_F32.
// A (16x4): lanes 0-15 hold M=lane, K={0,1}; lanes 16-31 hold M=lane-16, K={2,3}.
// B (4x16): lanes 0-15 hold N=lane, K={0,1}; lanes 16-31 hold N=lane-16, K={2,3}.
// C (16x16): VGPR v, lanes 0-15: (M=v, N=lane); lanes 16-31: (M=v+8, N=lane-16).
static __device__ __forceinline__ v8f wmma_k8(v2f a0, v2f a1, v2f b0, v2f b1, v8f acc) {
  acc = __builtin_amdgcn_wmma_f32_16x16x4_f32(false, a0, false, b0, (short)0, acc, false, false);
  acc = __builtin_amdgcn_wmma_f32_16x16x4_f32(false, a1, false, b1, (short)0, acc, false, false);
  return acc;
}

// One wave per (out-capsule j, 16-batch tile, i-chunk). Accumulates the partial
// tile s[d=16, b=16] over its 72 input capsules via WMMA; optionally scales the
// X fragment by routing weight c[b,i,j]. Writes raw partial to part[chunk][b][j][d].
template <bool WITH_C>
__global__ __launch_bounds__(32) void caps_gemm_partial(
    const float* __restrict__ X, const float* __restrict__ W,
    const float* __restrict__ Cw, float* __restrict__ part, int J) {
  const int j     = blockIdx.x / (4 * SPLITK);
  const int rem   = blockIdx.x % (4 * SPLITK);
  const int n0    = (rem >> 4) << 4;     // batch tile base: rem/16 in 0..3
  const int chunk = rem & (SPLITK - 1);  // i-chunk
  const int i0    = chunk * ICHUNK;
  const int lane  = threadIdx.x;
  const int half  = lane >> 4;
  const int lm    = lane & 15;
  const int bb    = n0 + lm;

  // W[i][j][d][k]: per-lane A ptr; step J*ND*NK per i.  X[b][i][k]: step NK per i.
  const float* wp = W + (((size_t)i0 * J + j) * ND + lm) * NK + 2 * half;
  const float* xp = X + ((size_t)bb * NI + i0) * NK + 2 * half;
  const size_t wstep = (size_t)J * ND * NK;

  v8f acc = {};
  for (int t = 0; t < ICHUNK; ++t) {
    __builtin_prefetch(wp + 8 * wstep, 0, 1);  // global_prefetch: W stream
    v2f a0 = *(const v2f*)(wp);
    v2f a1 = *(const v2f*)(wp + 4);
    v2f x0 = *(const v2f*)(xp);
    v2f x1 = *(const v2f*)(xp + 4);
    if (WITH_C) {
      float cw = Cw[((size_t)bb * NI + (size_t)(i0 + t)) * J + j];
      x0 *= cw;
      x1 *= cw;
    }
    acc = wmma_k8(a0, a1, x0, x1, acc);
    wp += wstep;
    xp += NK;
  }

  // part[chunk][b][j][d]: lane writes its 8 contiguous d values (d0 = 8*half)
  float* op = part + (((size_t)chunk * NB + bb) * J + j) * ND + 8 * half;
  v4f lo = {acc[0], acc[1], acc[2], acc[3]};
  v4f hi = {acc[4], acc[5], acc[6], acc[7]};
  *(v4f*)(op)     = lo;
  *(v4f*)(op + 4) = hi;
}

// Sum split-K partials, apply pre_scale (1/J on iteration 0), squash over d=16.
// One thread per (b, j); d is contiguous so loads vectorize.
__global__ void caps_reduce_squash(const float* __restrict__ part,
                                   float* __restrict__ dst, int J, float pre_scale) {
  int t = blockIdx.x * blockDim.x + threadIdx.x;
  if (t >= NB * J) return;
  const int b = t / J;
  const int j = t % J;

  float s[ND];
#pragma unroll
  for (int d = 0; d < ND; ++d) s[d] = 0.0f;
  for (int c = 0; c < SPLITK; ++c) {
    const float* p = part + (((size_t)c * NB + b) * J + j) * ND;
#pragma unroll
    for (int d = 0; d < ND; ++d) s[d] += p[d];
  }
  float ss = 0.0f;
#pragma unroll
  for (int d = 0; d < ND; ++d) {
    s[d] *= pre_scale;
    ss += s[d] * s[d];
  }
  // squash: (|s|^2/(1+|s|^2)) * s / sqrt(|s|^2 + eps)
  float fac = ss / ((1.0f + ss) * sqrtf(ss + EPSQ));
  float* o = dst + ((size_t)b * J + j) * ND;
#pragma unroll
  for (int d = 0; d < ND; ++d) o[d] = s[d] * fac;
}

// One wave per (16-batch tile, i); loops j. Recomputes P[d,b] tile via WMMA,
// dots with v0[b,j,:] (contraction over d = in-fragment), writes logits L[b,i,j].
__global__ __launch_bounds__(32) void caps_logits(
    const float* __restrict__ X, const float* __restrict__ W,
    const float* __restrict__ V0, float* __restrict__ L, int J) {
  const int i    = blockIdx.x >> 2;
  const int n0   = (blockIdx.x & 3) << 4;
  const int lane = threadIdx.x;
  const int half = lane >> 4;
  const int lm   = lane & 15;
  const int bb   = n0 + lm;

  const float* xp = X + ((size_t)bb * NI + (size_t)i) * NK + 2 * half;
  const v2f x0 = *(const v2f*)(xp);
  const v2f x1 = *(const v2f*)(xp + 4);

  for (int j = 0; j < J; ++j) {
    const float* wp = W + (((size_t)i * J + j) * ND + lm) * NK + 2 * half;
    v2f a0 = *(const v2f*)(wp);
    v2f a1 = *(const v2f*)(wp + 4);
    v8f p  = {};
    p = wmma_k8(a0, a1, x0, x1, p);

    const float* vp = V0 + ((size_t)bb * J + j) * ND + 8 * half;
    v4f vlo = *(const v4f*)(vp);
    v4f vhi = *(const v4f*)(vp + 4);
    float part = p[0] * vlo[0] + p[1] * vlo[1] + p[2] * vlo[2] + p[3] * vlo[3] +
                 p[4] * vhi[0] + p[5] * vhi[1] + p[6] * vhi[2] + p[7] * vhi[3];
    float tot = part + __shfl_xor(part, 16, 32);
    if (half == 0) L[((size_t)bb * NI + (size_t)i) * J + j] = tot;
  }
}

// In-place softmax over j for each (b,i) row of L.
__global__ void caps_softmax(float* __restrict__ L, int J) {
  int row = blockIdx.x * blockDim.x + threadIdx.x;
  if (row >= NB * NI) return;
  float* p = L + (size_t)row * J;
  float m = -3.402823466e38f;
  for (int j = 0; j < J; ++j) m = fmaxf(m, p[j]);
  float s = 0.0f;
  for (int j = 0; j < J; ++j) {
    float e = expf(p[j] - m);
    s += e;
    p[j] = e;
  }
  float inv = 1.0f / s;
  for (int j = 0; j < J; ++j) p[j] *= inv;
}

extern "C" void kernel_launch(void* const* d_in, const int* in_sizes, int n_in,
                              void* d_out, int out_size, void* d_ws, size_t ws_size,
                              hipStream_t stream) {
  (void)in_sizes; (void)n_in; (void)out_size; (void)ws_size;
  const float* X  = (const float*)d_in[0];   // [64,1152,8]
  const float* W0 = (const float*)d_in[1];   // [1152,20,16,8]
  const float* W1 = (const float*)d_in[2];   // [1152,100,16,8]
  float* out = (float*)d_out;                // [64,20,16] ++ [64,100,16]

  float* v0buf   = (float*)d_ws;                          // 64*100*16       (410 KB)
  float* cbuf    = v0buf + (size_t)NB * 100 * ND;         // 64*1152*100     (29.5 MB)
  float* partbuf = cbuf + (size_t)NB * NI * 100;          // 16*64*100*16    (6.6 MB)

  const float* Ws[2]   = {W0, W1};
  const int    Js[2]   = {20, 100};
  float*       Outs[2] = {out, out + (size_t)NB * 20 * ND};

  const dim3 wblk(32);
  for (int l = 0; l < 2; ++l) {
    const int J = Js[l];
    const dim3 ggrid(J * 4 * SPLITK);
    const dim3 rgrid((NB * J + 255) / 256), rblk(256);
    // it0: c uniform -> v0 = squash((1/J) * sum_i pred)
    caps_gemm_partial<false><<<ggrid, wblk, 0, stream>>>(X, Ws[l], nullptr, partbuf, J);
    caps_reduce_squash<<<rgrid, rblk, 0, stream>>>(partbuf, v0buf, J, 1.0f / (float)J);
    // agreement logits b[b,i,j] = <pred, v0>
    caps_logits<<<dim3(NI * 4), wblk, 0, stream>>>(X, Ws[l], v0buf, cbuf, J);
    // c = softmax_j(b)
    caps_softmax<<<dim3((NB * NI + 255) / 256), dim3(256), 0, stream>>>(cbuf, J);
    // it1: v = squash(sum_i c * pred)
    caps_gemm_partial<true><<<ggrid, wblk, 0, stream>>>(X, Ws[l], cbuf, partbuf, J);
    caps_reduce_squash<<<rgrid, rblk, 0, stream>>>(partbuf, Outs[l], J, 1.0f);
  }
}